// GAT_84086869721203
// MI455X (gfx1250) — compile-verified
//
#include <hip/hip_runtime.h>
#include <hip/hip_bf16.h>

#define NEG_SLOPE 0.2f
#define F_IN 14

typedef float v2f __attribute__((ext_vector_type(2)));
typedef float v8f __attribute__((ext_vector_type(8)));

// ---- monotonic float <-> uint encoding for atomic max on floats ----
__device__ __forceinline__ unsigned enc_f32(float f) {
    unsigned u = __float_as_uint(f);
    return (u & 0x80000000u) ? ~u : (u | 0x80000000u);
}
__device__ __forceinline__ float dec_f32(unsigned e) {
    unsigned u = (e & 0x80000000u) ? (e ^ 0x80000000u) : ~e;
    return __uint_as_float(u);
}

__device__ __forceinline__ void edge_sd(const int* __restrict__ ei, int E, int n,
                                        int idx, int& s, int& d) {
    if (idx < E) { s = ei[idx]; d = ei[E + idx]; }
    else         { s = idx - E; d = idx - E; }
    s = s < 0 ? 0 : (s >= n ? n - 1 : s);
    d = d < 0 ? 0 : (d >= n ? n - 1 : d);
}

// ============================================================
// GEMM 1: h1[n,64] = x[n,14] @ W1[14,64]  via V_WMMA_F32_16X16X4_F32
// one wave per 16-row tile; K padded 14->16 in LDS; 4 N-tiles of 16.
// ============================================================
__global__ __launch_bounds__(32) void k_gemm1(const float* __restrict__ x,
                                              const float* __restrict__ W1,
                                              float* __restrict__ h1, int n) {
    __shared__ float xs[16 * 16];    // padded x tile [row][k]
    __shared__ float ws[16 * 64];    // padded W1     [k][col]

    const int lane = threadIdx.x;
    const int row0 = blockIdx.x * 16;
    const bool hi  = lane >= 16;
    const int  m   = lane & 15;
    const bool full = (row0 + 16 <= n);   // workgroup-uniform

    // ---- stage W1 (zero-pad rows 14,15) ----
    for (int i = lane; i < 2 * 64 / 4; i += 32)          // 32 float4 of zeros
        ((float4*)(ws + 14 * 64))[i] = make_float4(0.f, 0.f, 0.f, 0.f);
    for (int i = lane; i < F_IN * 64 / 4; i += 32)       // 224 float4 copies
        ((float4*)ws)[i] = ((const float4*)W1)[i];

    // ---- stage x tile (zero-pad) ----
    for (int i = lane; i < 16 * 16 / 4; i += 32)
        ((float4*)xs)[i] = make_float4(0.f, 0.f, 0.f, 0.f);
    __syncthreads();
    {
        const float* xt = x + (size_t)row0 * F_IN;
        int rows = full ? 16 : (n - row0);
        int cnt  = rows * F_IN;                          // contiguous span
        for (int i = lane; i < cnt; i += 32) {
            int r = i / F_IN, c = i - r * F_IN;
            xs[r * 16 + c] = xt[i];
        }
    }
    __syncthreads();

    // ---- fragments ----
    const int kofs = hi ? 2 : 0;
    v2f a[4];
#pragma unroll
    for (int k = 0; k < 4; ++k) {
        a[k][0] = xs[m * 16 + 4 * k + kofs + 0];
        a[k][1] = xs[m * 16 + 4 * k + kofs + 1];
    }

#pragma unroll
    for (int t = 0; t < 4; ++t) {
        v8f c = {};
#pragma unroll
        for (int k = 0; k < 4; ++k) {
            v2f b;
            b[0] = ws[(4 * k + kofs + 0) * 64 + t * 16 + m];
            b[1] = ws[(4 * k + kofs + 1) * 64 + t * 16 + m];
            c = __builtin_amdgcn_wmma_f32_16x16x4_f32(false, a[k], false, b,
                                                      (short)0, c, false, false);
        }
        float* out = h1 + (size_t)row0 * 64 + t * 16 + m;
        if (full) {
#pragma unroll
            for (int r = 0; r < 8; ++r)
                out[(size_t)(r + (hi ? 8 : 0)) * 64] = c[r];
        } else {
#pragma unroll
            for (int r = 0; r < 8; ++r) {
                int rr = r + (hi ? 8 : 0);
                if (row0 + rr < n) out[(size_t)rr * 64] = c[r];
            }
        }
    }
}

// ============================================================
// GEMM 2: h2[n,8] = a1[n,64] @ W2[64,8]  (N padded 8->16, K=64 -> 16 wmma)
// ============================================================
__global__ __launch_bounds__(32) void k_gemm2(const float* __restrict__ a1,
                                              const float* __restrict__ W2,
                                              float* __restrict__ h2, int n) {
    __shared__ float smA[16 * 64];   // a1 tile [row][k]
    __shared__ float ws2[64 * 16];   // padded W2 [k][col], cols 8..15 zero

    const int lane = threadIdx.x;
    const int row0 = blockIdx.x * 16;
    const bool hi  = lane >= 16;
    const int  m   = lane & 15;
    const bool full = (row0 + 16 <= n);

    // ---- stage W2 zero-padded ----
    for (int i = lane; i < 64 * 16 / 4; i += 32)
        ((float4*)ws2)[i] = make_float4(0.f, 0.f, 0.f, 0.f);
    __syncthreads();
    for (int i = lane; i < 64 * 8 / 4; i += 32) {        // 128 float4 of W2
        int k = i >> 1, half = i & 1;
        ((float4*)(ws2 + k * 16 + half * 4))[0] = ((const float4*)W2)[i];
    }

    // ---- stage a1 tile ----
    if (full) {
        const float4* src = (const float4*)(a1 + (size_t)row0 * 64);
        for (int i = lane; i < 16 * 64 / 4; i += 32)
            ((float4*)smA)[i] = src[i];
    } else {
        for (int i = lane; i < 16 * 64 / 4; i += 32)
            ((float4*)smA)[i] = make_float4(0.f, 0.f, 0.f, 0.f);
        __syncthreads();
        int rows = n - row0;
        const float4* src = (const float4*)(a1 + (size_t)row0 * 64);
        for (int i = lane; i < rows * 16; i += 32)
            ((float4*)smA)[i] = src[i];
    }
    __syncthreads();

    const int kofs = hi ? 2 : 0;
    v8f c = {};
#pragma unroll
    for (int k = 0; k < 16; ++k) {
        v2f a, b;
        a[0] = smA[m * 64 + 4 * k + kofs + 0];
        a[1] = smA[m * 64 + 4 * k + kofs + 1];
        b[0] = ws2[(4 * k + kofs + 0) * 16 + m];
        b[1] = ws2[(4 * k + kofs + 1) * 16 + m];
        c = __builtin_amdgcn_wmma_f32_16x16x4_f32(false, a, false, b,
                                                  (short)0, c, false, false);
    }
    if (m < 8) {
        float* out = h2 + (size_t)row0 * 8 + m;
        if (full) {
#pragma unroll
            for (int r = 0; r < 8; ++r)
                out[(size_t)(r + (hi ? 8 : 0)) * 8] = c[r];
        } else {
#pragma unroll
            for (int r = 0; r < 8; ++r) {
                int rr = r + (hi ? 8 : 0);
                if (row0 + rr < n) out[(size_t)rr * 8] = c[r];
            }
        }
    }
}

// ============================================================
// per-node attention logits, layer 1: as1/ad1 [n,8]
// ============================================================
__global__ void k_alpha1(const float* __restrict__ h1,
                         const float* __restrict__ asrc, const float* __restrict__ adst,
                         float* __restrict__ as1, float* __restrict__ ad1, int n) {
    int tid = blockIdx.x * blockDim.x + threadIdx.x;
    if (tid >= n * 8) return;
    unsigned h = tid & 7u;
    const float4* hp = (const float4*)(h1 + (unsigned)(tid >> 3) * 64u + h * 8u);
    float4 v0 = hp[0], v1 = hp[1];
    const float4* sp = (const float4*)(asrc + h * 8u);
    const float4* dp = (const float4*)(adst + h * 8u);
    float4 s0 = sp[0], s1v = sp[1], d0 = dp[0], d1v = dp[1];
    float s = v0.x * s0.x + v0.y * s0.y + v0.z * s0.z + v0.w * s0.w
            + v1.x * s1v.x + v1.y * s1v.y + v1.z * s1v.z + v1.w * s1v.w;
    float d = v0.x * d0.x + v0.y * d0.y + v0.z * d0.z + v0.w * d0.w
            + v1.x * d1v.x + v1.y * d1v.y + v1.z * d1v.z + v1.w * d1v.w;
    as1[tid] = s;
    ad1[tid] = d;
}

// ============================================================
// layer-1 edge passes
// ============================================================
__global__ void k_edge_max1(const int* __restrict__ ei, int E, int n,
                            const float* __restrict__ as1, const float* __restrict__ ad1,
                            unsigned* __restrict__ m1) {
    int idx = blockIdx.x * blockDim.x + threadIdx.x;
    if (idx >= E + n) return;
    int s, d; edge_sd(ei, E, n, idx, s, d);
    const float4* ap = (const float4*)(as1 + (unsigned)s * 8u);
    const float4* bp = (const float4*)(ad1 + (unsigned)d * 8u);
    float4 a0 = ap[0], a1 = ap[1], b0 = bp[0], b1 = bp[1];
    float e[8] = {a0.x + b0.x, a0.y + b0.y, a0.z + b0.z, a0.w + b0.w,
                  a1.x + b1.x, a1.y + b1.y, a1.z + b1.z, a1.w + b1.w};
#pragma unroll
    for (int h = 0; h < 8; ++h) {
        float v = e[h] > 0.0f ? e[h] : NEG_SLOPE * e[h];
        atomicMax(&m1[(unsigned)d * 8u + h], enc_f32(v));
    }
}

__global__ void k_edge_sum1(const int* __restrict__ ei, int E, int n,
                            const float* __restrict__ as1, const float* __restrict__ ad1,
                            const unsigned* __restrict__ m1, float* __restrict__ s1) {
    int idx = blockIdx.x * blockDim.x + threadIdx.x;
    if (idx >= E + n) return;
    int s, d; edge_sd(ei, E, n, idx, s, d);
    const float4* ap = (const float4*)(as1 + (unsigned)s * 8u);
    const float4* bp = (const float4*)(ad1 + (unsigned)d * 8u);
    const uint4*  mp = (const uint4*)(m1 + (unsigned)d * 8u);
    float4 a0 = ap[0], a1 = ap[1], b0 = bp[0], b1 = bp[1];
    uint4  m0 = mp[0], m1v = mp[1];
    float e[8] = {a0.x + b0.x, a0.y + b0.y, a0.z + b0.z, a0.w + b0.w,
                  a1.x + b1.x, a1.y + b1.y, a1.z + b1.z, a1.w + b1.w};
    unsigned mm[8] = {m0.x, m0.y, m0.z, m0.w, m1v.x, m1v.y, m1v.z, m1v.w};
#pragma unroll
    for (int h = 0; h < 8; ++h) {
        float v = e[h] > 0.0f ? e[h] : NEG_SLOPE * e[h];
        atomicAdd(&s1[(unsigned)d * 8u + h], expf(v - dec_f32(mm[h])));
    }
}

__global__ void k_edge_scatter1(const int* __restrict__ ei, int E, int n,
                                const float* __restrict__ as1, const float* __restrict__ ad1,
                                const unsigned* __restrict__ m1, const float* __restrict__ s1,
                                const float* __restrict__ h1, float* __restrict__ acc1) {
    int tid = blockIdx.x * blockDim.x + threadIdx.x;
    if (tid >= (E + n) * 8) return;
    int idx = tid >> 3, h = tid & 7;
    int s, d; edge_sd(ei, E, n, idx, s, d);
    const float* hp = h1 + (unsigned)s * 64u + (unsigned)h * 8u;
    __builtin_prefetch(hp, 0, 3);                 // global_prefetch: hide gather latency
    float e = as1[(unsigned)s * 8u + h] + ad1[(unsigned)d * 8u + h];
    e = e > 0.0f ? e : NEG_SLOPE * e;
    float ex    = expf(e - dec_f32(m1[(unsigned)d * 8u + h]));
    float alpha = ex / (s1[(unsigned)d * 8u + h] + 1e-16f);
    float4 v0 = ((const float4*)hp)[0], v1 = ((const float4*)hp)[1];
    float* op = acc1 + (unsigned)d * 64u + (unsigned)h * 8u;
    atomicAdd(op + 0, v0.x * alpha);
    atomicAdd(op + 1, v0.y * alpha);
    atomicAdd(op + 2, v0.z * alpha);
    atomicAdd(op + 3, v0.w * alpha);
    atomicAdd(op + 4, v1.x * alpha);
    atomicAdd(op + 5, v1.y * alpha);
    atomicAdd(op + 6, v1.z * alpha);
    atomicAdd(op + 7, v1.w * alpha);
}

// bias + ELU, in place
__global__ void k_post1(float* __restrict__ acc1, const float* __restrict__ bias1, int n) {
    int tid = blockIdx.x * blockDim.x + threadIdx.x;
    if (tid >= n * 64) return;
    float v = acc1[tid] + bias1[tid & 63];
    acc1[tid] = v > 0.0f ? v : (expf(v) - 1.0f);
}

// ============================================================
// layer-2 (1 head, dim 8)
// ============================================================
__global__ void k_alpha2(const float* __restrict__ h2,
                         const float* __restrict__ asrc, const float* __restrict__ adst,
                         float* __restrict__ as2, float* __restrict__ ad2, int n) {
    int tid = blockIdx.x * blockDim.x + threadIdx.x;
    if (tid >= n) return;
    const float4* hp = (const float4*)(h2 + (unsigned)tid * 8u);
    float4 v0 = hp[0], v1 = hp[1];
    const float4* sp = (const float4*)asrc;
    const float4* dp = (const float4*)adst;
    float4 s0 = sp[0], s1v = sp[1], d0 = dp[0], d1v = dp[1];
    as2[tid] = v0.x * s0.x + v0.y * s0.y + v0.z * s0.z + v0.w * s0.w
             + v1.x * s1v.x + v1.y * s1v.y + v1.z * s1v.z + v1.w * s1v.w;
    ad2[tid] = v0.x * d0.x + v0.y * d0.y + v0.z * d0.z + v0.w * d0.w
             + v1.x * d1v.x + v1.y * d1v.y + v1.z * d1v.z + v1.w * d1v.w;
}

__global__ void k_edge_max2(const int* __restrict__ ei, int E, int n,
                            const float* __restrict__ as2, const float* __restrict__ ad2,
                            unsigned* __restrict__ m2) {
    int idx = blockIdx.x * blockDim.x + threadIdx.x;
    if (idx >= E + n) return;
    int s, d; edge_sd(ei, E, n, idx, s, d);
    float e = as2[s] + ad2[d];
    e = e > 0.0f ? e : NEG_SLOPE * e;
    atomicMax(&m2[d], enc_f32(e));
}

__global__ void k_edge_sum2(const int* __restrict__ ei, int E, int n,
                            const float* __restrict__ as2, const float* __restrict__ ad2,
                            const unsigned* __restrict__ m2, float* __restrict__ s2) {
    int idx = blockIdx.x * blockDim.x + threadIdx.x;
    if (idx >= E + n) return;
    int s, d; edge_sd(ei, E, n, idx, s, d);
    float e = as2[s] + ad2[d];
    e = e > 0.0f ? e : NEG_SLOPE * e;
    atomicAdd(&s2[d], expf(e - dec_f32(m2[d])));
}

__global__ void k_edge_scatter2(const int* __restrict__ ei, int E, int n,
                                const float* __restrict__ as2, const float* __restrict__ ad2,
                                const unsigned* __restrict__ m2, const float* __restrict__ s2,
                                const float* __restrict__ h2, float* __restrict__ out) {
    int idx = blockIdx.x * blockDim.x + threadIdx.x;
    if (idx >= E + n) return;
    int s, d; edge_sd(ei, E, n, idx, s, d);
    const float* hp = h2 + (unsigned)s * 8u;
    __builtin_prefetch(hp, 0, 3);
    float e = as2[s] + ad2[d];
    e = e > 0.0f ? e : NEG_SLOPE * e;
    float ex    = expf(e - dec_f32(m2[d]));
    float alpha = ex / (s2[d] + 1e-16f);
    float4 v0 = ((const float4*)hp)[0], v1 = ((const float4*)hp)[1];
    float* op = out + (unsigned)d * 8u;
    atomicAdd(op + 0, v0.x * alpha);
    atomicAdd(op + 1, v0.y * alpha);
    atomicAdd(op + 2, v0.z * alpha);
    atomicAdd(op + 3, v0.w * alpha);
    atomicAdd(op + 4, v1.x * alpha);
    atomicAdd(op + 5, v1.y * alpha);
    atomicAdd(op + 6, v1.z * alpha);
    atomicAdd(op + 7, v1.w * alpha);
}

__global__ void k_post2(float* __restrict__ out, const float* __restrict__ bias2, int n) {
    int tid = blockIdx.x * blockDim.x + threadIdx.x;
    if (tid >= n * 8) return;
    out[tid] += bias2[tid & 7];
}

// ============================================================
extern "C" void kernel_launch(void* const* d_in, const int* in_sizes, int n_in,
                              void* d_out, int out_size, void* d_ws, size_t ws_size,
                              hipStream_t stream) {
    const float* x     = (const float*)d_in[0];
    const int*   ei    = (const int*)d_in[1];
    const float* W1    = (const float*)d_in[2];
    const float* asrc1 = (const float*)d_in[3];
    const float* adst1 = (const float*)d_in[4];
    const float* bias1 = (const float*)d_in[5];
    const float* W2    = (const float*)d_in[6];
    const float* asrc2 = (const float*)d_in[7];
    const float* adst2 = (const float*)d_in[8];
    const float* bias2 = (const float*)d_in[9];

    int n = in_sizes[0] / F_IN;   // 100000
    int E = in_sizes[1] / 2;      // 1600000

    // ---- workspace carve (all segments 16B aligned) ----
    char* p = (char*)d_ws;
    float*    h1   = (float*)p;    p += (size_t)n * 64 * 4;
    float*    acc1 = (float*)p;    p += (size_t)n * 64 * 4;
    float*    as1  = (float*)p;    p += (size_t)n * 8 * 4;
    float*    ad1  = (float*)p;    p += (size_t)n * 8 * 4;
    float*    s1   = (float*)p;    p += (size_t)n * 8 * 4;
    unsigned* m1   = (unsigned*)p; p += (size_t)n * 8 * 4;
    float*    h2   = (float*)p;    p += (size_t)n * 8 * 4;
    float*    as2  = (float*)p;    p += (size_t)n * 4;
    float*    ad2  = (float*)p;    p += (size_t)n * 4;
    float*    s2   = (float*)p;    p += (size_t)n * 4;
    unsigned* m2   = (unsigned*)p; p += (size_t)n * 4;

    hipMemsetAsync(acc1, 0, (size_t)n * 64 * 4, stream);
    hipMemsetAsync(s1,   0, (size_t)n * 8 * 4, stream);
    hipMemsetAsync(m1,   0, (size_t)n * 8 * 4, stream);   // enc: 0 is below every real logit
    hipMemsetAsync(s2,   0, (size_t)n * 4, stream);
    hipMemsetAsync(m2,   0, (size_t)n * 4, stream);
    hipMemsetAsync(d_out, 0, (size_t)out_size * 4, stream);

    const int tpb = 256;
    int tot = E + n;
    int gEd = (tot + tpb - 1) / tpb;

    // ---- layer 1 ----
    k_gemm1<<<(n + 15) / 16, 32, 0, stream>>>(x, W1, h1, n);
    k_alpha1<<<(n * 8 + tpb - 1) / tpb, tpb, 0, stream>>>(h1, asrc1, adst1, as1, ad1, n);
    k_edge_max1<<<gEd, tpb, 0, stream>>>(ei, E, n, as1, ad1, m1);
    k_edge_sum1<<<gEd, tpb, 0, stream>>>(ei, E, n, as1, ad1, m1, s1);
    k_edge_scatter1<<<(tot * 8 + tpb - 1) / tpb, tpb, 0, stream>>>(ei, E, n, as1, ad1, m1, s1, h1, acc1);
    k_post1<<<(n * 64 + tpb - 1) / tpb, tpb, 0, stream>>>(acc1, bias1, n);

    // ---- layer 2 ----
    k_gemm2<<<(n + 15) / 16, 32, 0, stream>>>(acc1, W2, h2, n);
    k_alpha2<<<(n + tpb - 1) / tpb, tpb, 0, stream>>>(h2, asrc2, adst2, as2, ad2, n);
    k_edge_max2<<<gEd, tpb, 0, stream>>>(ei, E, n, as2, ad2, m2);
    k_edge_sum2<<<gEd, tpb, 0, stream>>>(ei, E, n, as2, ad2, m2, s2);
    k_edge_scatter2<<<gEd, tpb, 0, stream>>>(ei, E, n, as2, ad2, m2, s2, h2, (float*)d_out);
    k_post2<<<(n * 8 + tpb - 1) / tpb, tpb, 0, stream>>>((float*)d_out, bias2, n);
}